// ChoiceModelScalar_67723044323895
// MI455X (gfx1250) — compile-verified
//
#include <hip/hip_runtime.h>
#include <hip/hip_bf16.h>
#include <math.h>

#define MDIM   4096
#define HNUM   16
#define DH     16
#define DMODEL 32
#define BATCH  4096
#define NBIS   50

typedef __attribute__((ext_vector_type(2))) float v2f;
typedef __attribute__((ext_vector_type(8))) float v8f;

// ---------------------------------------------------------------------------
// Block reduction helpers (wave32: shuffle within wave, LDS across waves)
// ---------------------------------------------------------------------------
__device__ __forceinline__ float block_reduce_sum(float v, float* sm) {
    #pragma unroll
    for (int o = 16; o > 0; o >>= 1) v += __shfl_down(v, o, 32);
    const int lane = threadIdx.x & 31, w = threadIdx.x >> 5;
    const int nw = blockDim.x >> 5;
    if (lane == 0) sm[w] = v;
    __syncthreads();
    if (w == 0) {
        float r = (lane < nw) ? sm[lane] : 0.0f;
        #pragma unroll
        for (int o = 16; o > 0; o >>= 1) r += __shfl_down(r, o, 32);
        if (lane == 0) sm[0] = r;
    }
    __syncthreads();
    float res = sm[0];
    __syncthreads();
    return res;
}

__device__ __forceinline__ float block_reduce_max(float v, float* sm) {
    #pragma unroll
    for (int o = 16; o > 0; o >>= 1) v = fmaxf(v, __shfl_down(v, o, 32));
    const int lane = threadIdx.x & 31, w = threadIdx.x >> 5;
    const int nw = blockDim.x >> 5;
    if (lane == 0) sm[w] = v;
    __syncthreads();
    if (w == 0) {
        float r = (lane < nw) ? sm[lane] : -3.4e38f;
        #pragma unroll
        for (int o = 16; o > 0; o >>= 1) r = fmaxf(r, __shfl_down(r, o, 32));
        if (lane == 0) sm[0] = r;
    }
    __syncthreads();
    float res = sm[0];
    __syncthreads();
    return res;
}

// ---------------------------------------------------------------------------
// Kernel 1: embeddings + head projections.
//   A [H][M][DH]  (row-major per head, for WMMA A fragments)
//   Bt[H][DH][M]  (K-major per head, so WMMA B fragment loads are coalesced)
// ---------------------------------------------------------------------------
__global__ void embed_proj_kernel(const float* __restrict__ t_in,
                                  const float* __restrict__ prec_in,
                                  const float* __restrict__ W_pa,
                                  const float* __restrict__ W_pb,
                                  const float* __restrict__ Wa,
                                  const float* __restrict__ Wb,
                                  float* __restrict__ A,
                                  float* __restrict__ Bt) {
    __shared__ float sWa[HNUM * DH * DMODEL];
    __shared__ float sWb[HNUM * DH * DMODEL];
    for (int i = threadIdx.x; i < HNUM * DH * DMODEL; i += blockDim.x) {
        sWa[i] = Wa[i];
        sWb[i] = Wb[i];
    }
    __syncthreads();

    const int m = blockIdx.x * blockDim.x + threadIdx.x;
    const float prec = prec_in[m];
    const float tt   = t_in[m];

    float eA[DMODEL], eB[DMODEL];
    #pragma unroll
    for (int d = 0; d < DMODEL; ++d) {
        const float pv   = __powf(10000.0f, 2.0f * (float)(d >> 1) / (float)DMODEL);
        const float ph_p = prec / pv;
        const float ph_t = tt / pv;
        const float pe_sp = ((d & 1) == 0) ? __sinf(ph_p) : __cosf(ph_p);
        const float pe_t  = ((d & 1) == 0) ? __sinf(ph_t) : __cosf(ph_t);
        eA[d] = pe_sp + prec * W_pa[d] + pe_t;
        eB[d] = pe_sp + prec * W_pb[d] + pe_t;
    }

    for (int h = 0; h < HNUM; ++h) {
        #pragma unroll
        for (int k = 0; k < DH; ++k) {
            const float* wa = &sWa[(h * DH + k) * DMODEL];
            const float* wb = &sWb[(h * DH + k) * DMODEL];
            float accA = 0.0f, accB = 0.0f;
            #pragma unroll
            for (int d = 0; d < DMODEL; ++d) {
                accA = fmaf(eA[d], wa[d], accA);
                accB = fmaf(eB[d], wb[d], accB);
            }
            A [((size_t)(h * MDIM + m)) * DH + k] = accA;
            Bt[((size_t)(h * DH + k)) * MDIM + m] = accB;
        }
    }
}

// ---------------------------------------------------------------------------
// Kernel 2: matrix_H[h] = A_H[h] (4096x16) @ B_H[h]^T (16x4096) via
// V_WMMA_F32_16X16X4_F32, K=16 as four chained x4 WMMAs. Store-BW bound
// (1.07 GB out) -> nontemporal stores. One block = (h, mi); 8 waves split ni.
// ---------------------------------------------------------------------------
__global__ void gemm_wmma_kernel(const float* __restrict__ A,
                                 const float* __restrict__ Bt,
                                 float* __restrict__ outM) {
    const int bid  = blockIdx.x;
    const int h    = bid >> 8;          // 256 mi-tiles per head
    const int mi   = bid & 255;
    const int wave = threadIdx.x >> 5;
    const int lane = threadIdx.x & 31;
    const int half = lane >> 4;         // 0 or 1
    const int lrow = lane & 15;

    // A fragment per k-chunk: a[j] = A[row=lrow][k = koff + j + 2*half]
    const float* Ahm = A + ((size_t)(h * MDIM + mi * 16 + lrow)) * DH;
    v2f aF[4];
    #pragma unroll
    for (int kk = 0; kk < 4; ++kk) {
        aF[kk][0] = Ahm[kk * 4 + 2 * half + 0];
        aF[kk][1] = Ahm[kk * 4 + 2 * half + 1];
    }

    const float* Bth = Bt + (size_t)h * DH * MDIM;

    for (int ni = wave; ni < MDIM / 16; ni += 8) {
        const int ncol = ni * 16 + lrow;
        v8f acc = {};
        #pragma unroll
        for (int kk = 0; kk < 4; ++kk) {
            // B fragment: b[j] = Bt[k = koff + j + 2*half][ncol]
            v2f bF;
            const int k0 = kk * 4 + 2 * half;
            bF[0] = Bth[(size_t)(k0 + 0) * MDIM + ncol];
            bF[1] = Bth[(size_t)(k0 + 1) * MDIM + ncol];
            acc = __builtin_amdgcn_wmma_f32_16x16x4_f32(
                false, aF[kk], false, bF, (short)0, acc, false, false);
        }
        // D layout: vgpr j -> (m = mi*16 + j + 8*half, n = ncol)
        float* op = outM + ((size_t)(h * MDIM + mi * 16 + 8 * half)) * MDIM + ncol;
        #pragma unroll
        for (int j = 0; j < 8; ++j) {
            __builtin_nontemporal_store(acc[j], op + (size_t)j * MDIM);
        }
    }
}

// ---------------------------------------------------------------------------
// Kernel 3: sumA[h][k] = sum_m A[h][m][k]   (z_H = sumA . B_H without
// re-reading the 1 GB matrix)
// ---------------------------------------------------------------------------
__global__ void colsum_kernel(const float* __restrict__ A,
                              float* __restrict__ sumA) {
    __shared__ float sm[256];
    const int h = blockIdx.x;
    const int t = threadIdx.x;
    const int k = t & 15;
    const int s = t >> 4;
    float acc = 0.0f;
    for (int m = s; m < MDIM; m += 16)
        acc += A[((size_t)(h * MDIM + m)) * DH + k];
    sm[t] = acc;
    __syncthreads();
    for (int off = 128; off >= 16; off >>= 1) {
        if (t < off) sm[t] += sm[t + off];
        __syncthreads();
    }
    if (t < 16) sumA[h * DH + t] = sm[t];
}

// ---------------------------------------------------------------------------
// Kernel 4: per-head z, normalize, entmax bisection (50 iters), g_res + lower.
// One 1024-thread block per head; each thread owns 4 of the 4096 columns.
// ---------------------------------------------------------------------------
__global__ void entmax_kernel(const float* __restrict__ Bt,
                              const float* __restrict__ sumA,
                              const float* __restrict__ alpha,
                              const float* __restrict__ U,
                              float* __restrict__ g_out,   // [H][M]
                              float* __restrict__ lower) { // [H]
    __shared__ float sm[32];
    const int h   = blockIdx.x;
    const int tid = threadIdx.x;

    float sA[DH];
    #pragma unroll
    for (int k = 0; k < DH; ++k) sA[k] = sumA[h * DH + k];

    // z[h][n] = sumA[h] . B_H[h][n]
    float z[4];
    #pragma unroll
    for (int i = 0; i < 4; ++i) {
        const int n = tid + i * 1024;
        float acc = 0.0f;
        #pragma unroll
        for (int k = 0; k < DH; ++k)
            acc = fmaf(sA[k], Bt[((size_t)(h * DH + k)) * MDIM + n], acc);
        z[i] = acc;
    }

    float ss = 0.0f;
    #pragma unroll
    for (int i = 0; i < 4; ++i) ss += z[i] * z[i];
    ss = block_reduce_sum(ss, sm);
    const float norm = fmaxf(sqrtf(ss), 1e-12f);

    const float am1 = alpha[h] - 1.0f;
    const float inv = 1.0f / am1;
    float xs[4];
    #pragma unroll
    for (int i = 0; i < 4; ++i) xs[i] = (z[i] / norm) * am1;

    float mx = xs[0];
    #pragma unroll
    for (int i = 1; i < 4; ++i) mx = fmaxf(mx, xs[i]);
    mx = block_reduce_max(mx, sm);

    float tau_lo = mx - 1.0f;
    const float tau_hi = mx - powf(1.0f / (float)MDIM, am1);

    float p[4];
    float fl = 0.0f;
    #pragma unroll
    for (int i = 0; i < 4; ++i) {
        p[i] = powf(fmaxf(xs[i] - tau_lo, 0.0f), inv);
        fl += p[i];
    }
    const float f_lo = block_reduce_sum(fl, sm) - 1.0f;

    float dm = tau_hi - tau_lo;
    for (int it = 0; it < NBIS; ++it) {
        dm *= 0.5f;
        const float tau_m = tau_lo + dm;
        float s = 0.0f;
        #pragma unroll
        for (int i = 0; i < 4; ++i) {
            p[i] = powf(fmaxf(xs[i] - tau_m, 0.0f), inv);
            s += p[i];
        }
        const float f_m = block_reduce_sum(s, sm) - 1.0f;   // block-uniform
        if (f_m * f_lo >= 0.0f) tau_lo = tau_m;
    }

    float sp = 0.0f;
    #pragma unroll
    for (int i = 0; i < 4; ++i) sp += p[i];
    sp = block_reduce_sum(sp, sm);

    float gsum = 0.0f;
    #pragma unroll
    for (int i = 0; i < 4; ++i) {
        const int n = tid + i * 1024;
        const float g = p[i] / sp;
        g_out[(size_t)h * MDIM + n] = g;
        gsum += g * expf(U[(size_t)h * MDIM + n]);
    }
    gsum = block_reduce_sum(gsum, sm);
    if (tid == 0) lower[h] = gsum;
}

// ---------------------------------------------------------------------------
// Kernel 5: P_M[n] = sum_h softmax(pi)[h] * max(g*exp(U),0) / lower[h]
// ---------------------------------------------------------------------------
__global__ void mixture_kernel(const float* __restrict__ g_out,
                               const float* __restrict__ U,
                               const float* __restrict__ pi,
                               const float* __restrict__ lower,
                               float* __restrict__ PM) {
    const int n = blockIdx.x * blockDim.x + threadIdx.x;
    float pmax = -3.4e38f;
    #pragma unroll
    for (int h = 0; h < HNUM; ++h) pmax = fmaxf(pmax, pi[h]);
    float wv[HNUM];
    float wsum = 0.0f;
    #pragma unroll
    for (int h = 0; h < HNUM; ++h) { wv[h] = expf(pi[h] - pmax); wsum += wv[h]; }

    float P = 0.0f;
    #pragma unroll
    for (int h = 0; h < HNUM; ++h) {
        const float g  = g_out[(size_t)h * MDIM + n];
        const float gu = g * expf(U[(size_t)h * MDIM + n]);
        P += (wv[h] / wsum) * fmaxf(gu, 0.0f) / lower[h];
    }
    PM[n] = P;
}

// ---------------------------------------------------------------------------
// Kernel 6: batch_sum_ll = sum_b log(P_M[indices[b]])
// ---------------------------------------------------------------------------
__global__ void loglik_kernel(const float* __restrict__ PM,
                              const int* __restrict__ indices,
                              float* __restrict__ out0) {
    __shared__ float sm[32];
    const int tid = threadIdx.x;
    float s = 0.0f;
    #pragma unroll
    for (int i = 0; i < BATCH / 1024; ++i)
        s += logf(PM[indices[tid + i * 1024]]);
    s = block_reduce_sum(s, sm);
    if (tid == 0) out0[0] = s;
}

// ---------------------------------------------------------------------------
extern "C" void kernel_launch(void* const* d_in, const int* in_sizes, int n_in,
                              void* d_out, int out_size, void* d_ws, size_t ws_size,
                              hipStream_t stream) {
    const float* unique_t = (const float*)d_in[0];
    const float* prec     = (const float*)d_in[1];
    const float* W_pa     = (const float*)d_in[2];
    const float* W_pb     = (const float*)d_in[3];
    const float* Wa       = (const float*)d_in[4];
    const float* Wb       = (const float*)d_in[5];
    const float* alpha    = (const float*)d_in[6];
    const float* U        = (const float*)d_in[7];
    const float* pi       = (const float*)d_in[8];
    const int*   indices  = (const int*)d_in[9];

    float* out = (float*)d_out;
    // output layout: [0] = batch_sum_ll, [1 ..] = matrix_H (H*M*M), then g_res_H (H*M)
    const size_t GOFF = 1 + (size_t)HNUM * MDIM * MDIM;

    float* ws    = (float*)d_ws;
    float* A     = ws;                                  // H*M*DH   = 1,048,576 f
    float* Bt    = A + (size_t)HNUM * MDIM * DH;        // H*DH*M   = 1,048,576 f
    float* sumA  = Bt + (size_t)HNUM * DH * MDIM;       // 256 f
    float* lower = sumA + HNUM * DH;                    // 16 f
    float* PM    = lower + HNUM;                        // 4096 f

    embed_proj_kernel<<<MDIM / 256, 256, 0, stream>>>(unique_t, prec, W_pa, W_pb,
                                                      Wa, Wb, A, Bt);
    gemm_wmma_kernel<<<HNUM * (MDIM / 16), 256, 0, stream>>>(A, Bt, out + 1);
    colsum_kernel<<<HNUM, 256, 0, stream>>>(A, sumA);
    entmax_kernel<<<HNUM, 1024, 0, stream>>>(Bt, sumA, alpha, U, out + GOFF, lower);
    mixture_kernel<<<MDIM / 256, 256, 0, stream>>>(out + GOFF, U, pi, lower, PM);
    loglik_kernel<<<1, 1024, 0, stream>>>(PM, indices, out);
}